// SumModel_19490561589963
// MI455X (gfx1250) — compile-verified
//
#include <hip/hip_runtime.h>
#include <stdint.h>

// ---------------------------------------------------------------------------
// Single-pass decoupled-lookback cumsum over [B=128][NITER=8192][C=64].
//
// Roofline: 256 MB read + 256 MB write = 512 MB @ 23.3 TB/s  =>  ~22 us floor.
// x/y are touched exactly once (plus ~4 MB of lookback slots), so the kernel
// sits on the HBM floor. Tiles are staged through LDS with CDNA5 async
// global<->LDS b128 transfers (ASYNCcnt) for wide, register-free data
// movement. Streaming traffic is marked non-temporal (TH_*_NT) so the 512 MB
// stream does not wash the 192 MB L2; the hot lookback slots stay cached.
// ---------------------------------------------------------------------------

#define NITER      8192
#define CCH        64          // channels
#define TILE_IT    128         // iterations per tile
#define NT         (NITER / TILE_IT)   // 64 tiles per row
#define LOG_NT     6
#define BLOCK      256
#define JG         (BLOCK / CCH)       // 4 sub-chunks of iterations
#define SPT        (TILE_IT / JG)      // 32 iterations per thread
#define TILE_FLOATS (TILE_IT * CCH)    // 8192 floats = 32 KB
#define ROUNDS     (TILE_FLOATS / (BLOCK * 4))  // 8 async b128 rounds

// ---- CDNA5 async LDS transfers (VGLOBAL async ops, tracked by ASYNCcnt) ----
__device__ __forceinline__ void async_load_b128_nt(uint32_t lds_byte_off,
                                                   const void* gaddr) {
  asm volatile("global_load_async_to_lds_b128 %0, %1, off th:TH_LOAD_NT"
               :: "v"(lds_byte_off), "v"((uint64_t)(uintptr_t)gaddr)
               : "memory");
}
__device__ __forceinline__ void async_store_b128_nt(void* gaddr,
                                                    uint32_t lds_byte_off) {
  asm volatile("global_store_async_from_lds_b128 %0, %1, off th:TH_STORE_NT"
               :: "v"((uint64_t)(uintptr_t)gaddr), "v"(lds_byte_off)
               : "memory");
}
__device__ __forceinline__ void wait_asynccnt0() {
  asm volatile("s_wait_asynccnt 0" ::: "memory");
}

// Zero lookback slots each call (harness poisons ws once, never re-poisons).
__global__ void init_slots_kernel(unsigned long long* slots, size_t n) {
  size_t i = (size_t)blockIdx.x * blockDim.x + threadIdx.x;
  if (i < n) slots[i] = 0ull;
}

// Packed lookback slot: bits[63:32] = status (0 none, 1 aggregate, 2 inclusive)
//                       bits[31:0]  = float payload bits
__global__ __launch_bounds__(BLOCK)
void scan_lookback_kernel(const float* __restrict__ x,
                          const float* __restrict__ s_init,
                          float* __restrict__ y,
                          unsigned long long* __restrict__ slots) {
  __shared__ __align__(16) float tile[TILE_FLOATS];   // 32 KB
  __shared__ float excl_intra[JG * CCH];              // 1 KB

  const int blk = blockIdx.x;
  const int t   = blk & (NT - 1);     // tile index within row (ascending first)
  const int b   = blk >> LOG_NT;      // batch row
  const int tid = threadIdx.x;

  const size_t rowbase = (size_t)b * (size_t)(NITER * CCH);
  const float* gsrc = x + rowbase + (size_t)t * TILE_FLOATS;
  float*       gdst = y + rowbase + (size_t)t * TILE_FLOATS;

  const uint32_t lds_base = (uint32_t)(uintptr_t)tile;

  // ---- Phase 1: async-stage the contiguous 32 KB tile into LDS ----
#pragma unroll
  for (int r = 0; r < ROUNDS; ++r) {
    const int fo = (r * BLOCK + tid) * 4;           // float offset (16B/lane)
    async_load_b128_nt(lds_base + (uint32_t)fo * 4u, gsrc + fo);
  }
  wait_asynccnt0();
  __syncthreads();

  const int c = tid & (CCH - 1);
  const int j = tid >> 6;

  // ---- Phase 2: per-thread partial sums over SPT iterations ----
  float tot = 0.f;
#pragma unroll
  for (int k = 0; k < SPT; ++k)
    tot += tile[(j * SPT + k) * CCH + c];           // bank = c, conflict-free
  excl_intra[j * CCH + c] = tot;
  __syncthreads();

  // ---- Phase 3: channel threads (tid<64) scan sub-chunks + lookback ----
  if (tid < CCH) {
    float run = 0.f;
#pragma unroll
    for (int jj = 0; jj < JG; ++jj) {
      float v = excl_intra[jj * CCH + tid];
      excl_intra[jj * CCH + tid] = run;             // exclusive within tile
      run += v;
    }
    const float agg = run;                          // tile total, this channel

    unsigned long long* myslots = slots + ((size_t)b * NT) * CCH;

    if (t > 0) {                                    // publish aggregate ASAP
      unsigned long long pub =
          (1ull << 32) | (unsigned long long)__float_as_uint(agg);
      __hip_atomic_store(&myslots[(size_t)t * CCH + tid], pub,
                         __ATOMIC_RELEASE, __HIP_MEMORY_SCOPE_AGENT);
    }

    // decoupled lookback over predecessor tiles of this (b, c) chain
    float excl = 0.f;
    for (int look = t - 1; look >= 0; --look) {
      unsigned long long v;
      do {
        v = __hip_atomic_load(&myslots[(size_t)look * CCH + tid],
                              __ATOMIC_ACQUIRE, __HIP_MEMORY_SCOPE_AGENT);
      } while ((v >> 32) == 0ull);
      excl += __uint_as_float((uint32_t)(v & 0xffffffffu));
      if ((v >> 32) == 2ull) break;                 // found an inclusive prefix
    }

    // publish inclusive prefix for successors
    unsigned long long pub2 =
        (2ull << 32) | (unsigned long long)__float_as_uint(excl + agg);
    __hip_atomic_store(&myslots[(size_t)t * CCH + tid], pub2,
                       __ATOMIC_RELEASE, __HIP_MEMORY_SCOPE_AGENT);

    const float comb = excl + s_init[tid];          // fold s_init in once
#pragma unroll
    for (int jj = 0; jj < JG; ++jj)
      excl_intra[jj * CCH + tid] += comb;
  }
  __syncthreads();

  // ---- Phase 4: finalize running sums in-place in LDS ----
  float run = excl_intra[j * CCH + c];
#pragma unroll
  for (int k = 0; k < SPT; ++k) {
    const int idx = (j * SPT + k) * CCH + c;
    run += tile[idx];
    tile[idx] = run;
  }
  __syncthreads();

  // ---- Phase 5: async-stream the finished 32 KB tile out of LDS ----
#pragma unroll
  for (int r = 0; r < ROUNDS; ++r) {
    const int fo = (r * BLOCK + tid) * 4;
    async_store_b128_nt(gdst + fo, lds_base + (uint32_t)fo * 4u);
  }
  wait_asynccnt0();
}

// Fallback (only if ws_size is unexpectedly tiny): one thread per chain.
__global__ void naive_scan_kernel(const float* __restrict__ x,
                                  const float* __restrict__ s_init,
                                  float* __restrict__ y) {
  const int b = blockIdx.x;
  const int c = threadIdx.x;
  const size_t base = (size_t)b * (size_t)(NITER * CCH) + c;
  float s = s_init[c];
  for (int i = 0; i < NITER; ++i) {
    s += x[base + (size_t)i * CCH];
    y[base + (size_t)i * CCH] = s;
  }
}

extern "C" void kernel_launch(void* const* d_in, const int* in_sizes, int n_in,
                              void* d_out, int out_size, void* d_ws,
                              size_t ws_size, hipStream_t stream) {
  const float* x      = (const float*)d_in[0];
  const float* s_init = (const float*)d_in[1];
  float*       y      = (float*)d_out;

  const int nrows = in_sizes[0] / (NITER * CCH);           // 128
  const size_t nslots = (size_t)nrows * NT * CCH;          // 128*64*64 = 512K
  const size_t need   = nslots * sizeof(unsigned long long); // 4 MB

  if (ws_size >= need) {
    unsigned long long* slots = (unsigned long long*)d_ws;
    init_slots_kernel<<<(unsigned)((nslots + 255) / 256), 256, 0, stream>>>(
        slots, nslots);
    scan_lookback_kernel<<<(unsigned)(nrows * NT), BLOCK, 0, stream>>>(
        x, s_init, y, slots);
  } else {
    naive_scan_kernel<<<(unsigned)nrows, CCH, 0, stream>>>(x, s_init, y);
  }
}